// Attention_55027120996971
// MI455X (gfx1250) — compile-verified
//
#include <hip/hip_runtime.h>
#include <hip/hip_bf16.h>

// MI455X / gfx1250: fp32 attention forward built on V_WMMA_F32_16X16X4_F32.
// Compute-bound problem (~69 GFLOP vs ~83MB -> 3.6us at 23.3TB/s), fp32
// outputs (k,v validated directly) => native fp32 matrix path, no conversion.

typedef float v2f __attribute__((ext_vector_type(2)));
typedef float v8f __attribute__((ext_vector_type(8)));

#define D_MODEL 1024
#define T_SEQ   2048
#define BATCH   2
#define HEADS   16
#define DK      64
#define NEG_BIG (-1e30f)

static __device__ __forceinline__ v8f wmma_f32(v2f a, v2f b, v8f c) {
  // D = A(16x4) * B(4x16) + C(16x16), fp32, wave32.
  return __builtin_amdgcn_wmma_f32_16x16x4_f32(false, a, false, b,
                                               (short)0, c, false, false);
}

// ---------------------------------------------------------------------------
// Kernel 1: qkv = x @ w_qkv^T + b_qkv ; scatter into q(ws), k(out), v(out)
// in [B, H, T, dk] layout. M=4096, N=3072, K=1024.
// Wave tile: 16(M) x 64(N). Block = 8 waves covering 16 x 512.
// ---------------------------------------------------------------------------
__global__ __launch_bounds__(256) void qkv_proj_kernel(
    const float* __restrict__ x, const float* __restrict__ w,
    const float* __restrict__ bias,
    float* __restrict__ qws, float* __restrict__ kout, float* __restrict__ vout)
{
  const int lane = threadIdx.x & 31;
  const int wv   = threadIdx.x >> 5;
  const int half = lane >> 4;
  const int row  = lane & 15;
  const int m0    = blockIdx.y * 16;
  const int nbase = blockIdx.x * 512 + wv * 64;

  v8f c[4];
  #pragma unroll
  for (int j = 0; j < 4; ++j) c[j] = v8f{};

  const float* xrow = x + (size_t)(m0 + row) * D_MODEL + 2 * half;
  const float* wrow = w + (size_t)(nbase + row) * D_MODEL + 2 * half;

  #pragma unroll 4
  for (int k0 = 0; k0 < D_MODEL; k0 += 4) {
    if ((k0 & 63) == 0 && k0 + 256 < D_MODEL) {
      __builtin_prefetch(xrow + k0 + 256, 0, 1);
      __builtin_prefetch(wrow + k0 + 256, 0, 1);
    }
    v2f a = *(const v2f*)(xrow + k0);
    #pragma unroll
    for (int j = 0; j < 4; ++j) {
      v2f b = *(const v2f*)(wrow + (size_t)j * 16 * D_MODEL + k0);
      c[j] = wmma_f32(a, b, c[j]);
    }
  }

  #pragma unroll
  for (int j = 0; j < 4; ++j) {
    const int n    = nbase + 16 * j + row;      // column in [0, 3072)
    const int sect = n >> 10;                   // 0=q, 1=k, 2=v
    const int d    = n & (D_MODEL - 1);
    const int h    = d >> 6;
    const int dki  = d & (DK - 1);
    const float bv = bias[n];
    float* dst = (sect == 0) ? qws : (sect == 1) ? kout : vout;
    #pragma unroll
    for (int r = 0; r < 8; ++r) {
      const int m = m0 + r + 8 * half;          // token row in [0, 4096)
      const int b = m >> 11;
      const int t = m & (T_SEQ - 1);
      dst[((((size_t)b * HEADS + h) * T_SEQ) + t) * DK + dki] = c[j][r] + bv;
    }
  }
}

// ---------------------------------------------------------------------------
// Kernel 2: flash-style causal attention per (b,h). One wave owns 16 q-rows,
// block = 4 waves (64 q-rows). Online softmax fully in registers; P is staged
// through wave-private LDS to become the A operand of P@V.
// ---------------------------------------------------------------------------
__global__ __launch_bounds__(128) void attn_kernel(
    const float* __restrict__ q, const float* __restrict__ kmat,
    const float* __restrict__ vmat, float* __restrict__ ctx)
{
  __shared__ __align__(16) float sP[4][16 * 16];

  const int lane = threadIdx.x & 31;
  const int wv   = threadIdx.x >> 5;
  const int half = lane >> 4;
  const int row  = lane & 15;
  const int bh   = blockIdx.y;
  const int b    = bh >> 4;
  const int h    = bh & (HEADS - 1);
  const int qrow0 = blockIdx.x * 64 + wv * 16;

  const float* qp = q    + (size_t)bh * T_SEQ * DK;
  const float* kp = kmat + (size_t)bh * T_SEQ * DK;
  const float* vp = vmat + (size_t)bh * T_SEQ * DK;

  // Register-resident Q fragments: 16 k-steps of 4 over dk=64.
  v2f aq[16];
  #pragma unroll
  for (int kk = 0; kk < 16; ++kk)
    aq[kk] = *(const v2f*)(qp + (size_t)(qrow0 + row) * DK + 4 * kk + 2 * half);

  v8f acc[4];
  float mrun[8], lrun[8];
  #pragma unroll
  for (int j = 0; j < 4; ++j) acc[j] = v8f{};
  #pragma unroll
  for (int r = 0; r < 8; ++r) { mrun[r] = -3.0e38f; lrun[r] = 0.0f; }

  float* sPw = sP[wv];
  const int nkt = qrow0 / 16 + 1;               // causal extent in 16-col tiles

  for (int kt = 0; kt < nkt; ++kt) {
    // ---- S = (Q K^T) tile, 16x16, 16 wmma steps over dk ----
    v8f s = v8f{};
    const float* kb = kp + (size_t)(kt * 16 + row) * DK + 2 * half;
    #pragma unroll
    for (int kk = 0; kk < 16; ++kk)
      s = wmma_f32(aq[kk], *(const v2f*)(kb + 4 * kk), s);

    // ---- scale + causal mask (element r: qrow = qrow0+r+8*half, col below)
    const int col = kt * 16 + row;
    #pragma unroll
    for (int r = 0; r < 8; ++r) {
      const int qg = qrow0 + r + 8 * half;
      const float val = s[r] * 0.125f;          // 1/sqrt(dk)
      s[r] = (col <= qg) ? val : NEG_BIG;
    }

    // ---- online softmax: row stats via shfl_xor within 16-lane half-group
    float p[8];
    #pragma unroll
    for (int r = 0; r < 8; ++r) {
      float tm = s[r];
      tm = fmaxf(tm, __shfl_xor(tm, 1));
      tm = fmaxf(tm, __shfl_xor(tm, 2));
      tm = fmaxf(tm, __shfl_xor(tm, 4));
      tm = fmaxf(tm, __shfl_xor(tm, 8));
      const float mnew  = fmaxf(mrun[r], tm);
      const float alpha = __expf(mrun[r] - mnew);
      const float pe    = __expf(s[r] - mnew);  // masked -> exp(-huge) = 0
      float ls = pe;
      ls += __shfl_xor(ls, 1);
      ls += __shfl_xor(ls, 2);
      ls += __shfl_xor(ls, 4);
      ls += __shfl_xor(ls, 8);
      lrun[r] = lrun[r] * alpha + ls;
      mrun[r] = mnew;
      p[r] = pe;
      #pragma unroll
      for (int j = 0; j < 4; ++j) acc[j][r] *= alpha;
    }

    // ---- stage P (C layout -> row-major LDS tile), wave-private region ----
    #pragma unroll
    for (int r = 0; r < 8; ++r)
      sPw[(r + 8 * half) * 16 + row] = p[r];
    asm volatile("s_wait_dscnt 0" ::: "memory");

    // ---- acc += P(16x16) @ V(16x64): 4 k-steps x 4 n-tiles ----
    #pragma unroll
    for (int kk2 = 0; kk2 < 4; ++kk2) {
      v2f a = *(const v2f*)(&sPw[row * 16 + 4 * kk2 + 2 * half]);
      const float* vb = vp + (size_t)(kt * 16 + 4 * kk2 + 2 * half) * DK + row;
      #pragma unroll
      for (int j = 0; j < 4; ++j) {
        v2f bfr;
        bfr.x = vb[j * 16];
        bfr.y = vb[DK + j * 16];
        acc[j] = wmma_f32(a, bfr, acc[j]);
      }
    }
    asm volatile("s_wait_dscnt 0" ::: "memory"); // reads done before next overwrite
  }

  // ---- normalize and write ctx in [B, T, H*dk] (= concat) layout ----
  float* op = ctx + (size_t)b * T_SEQ * D_MODEL + (size_t)h * DK;
  #pragma unroll
  for (int r = 0; r < 8; ++r) {
    const float inv = 1.0f / lrun[r];
    const int t = qrow0 + r + 8 * half;
    #pragma unroll
    for (int j = 0; j < 4; ++j)
      op[(size_t)t * D_MODEL + j * 16 + row] = acc[j][r] * inv;
  }
}

// ---------------------------------------------------------------------------
// Kernel 3: out = ctx @ w_o^T + b_o. M=4096, N=1024, K=1024.
// ---------------------------------------------------------------------------
__global__ __launch_bounds__(256) void out_proj_kernel(
    const float* __restrict__ ctx, const float* __restrict__ w,
    const float* __restrict__ bias, float* __restrict__ out)
{
  const int lane = threadIdx.x & 31;
  const int wv   = threadIdx.x >> 5;
  const int half = lane >> 4;
  const int row  = lane & 15;
  const int m0    = blockIdx.y * 16;
  const int nbase = blockIdx.x * 512 + wv * 64;

  v8f c[4];
  #pragma unroll
  for (int j = 0; j < 4; ++j) c[j] = v8f{};

  const float* arow = ctx + (size_t)(m0 + row) * D_MODEL + 2 * half;
  const float* wrow = w   + (size_t)(nbase + row) * D_MODEL + 2 * half;

  #pragma unroll 4
  for (int k0 = 0; k0 < D_MODEL; k0 += 4) {
    v2f a = *(const v2f*)(arow + k0);
    #pragma unroll
    for (int j = 0; j < 4; ++j) {
      v2f b = *(const v2f*)(wrow + (size_t)j * 16 * D_MODEL + k0);
      c[j] = wmma_f32(a, b, c[j]);
    }
  }

  #pragma unroll
  for (int j = 0; j < 4; ++j) {
    const int n    = nbase + 16 * j + row;
    const float bv = bias[n];
    #pragma unroll
    for (int r = 0; r < 8; ++r) {
      const int m = m0 + r + 8 * half;
      out[(size_t)m * D_MODEL + n] = c[j][r] + bv;
    }
  }
}

// ---------------------------------------------------------------------------
extern "C" void kernel_launch(void* const* d_in, const int* in_sizes, int n_in,
                              void* d_out, int out_size, void* d_ws, size_t ws_size,
                              hipStream_t stream) {
  const float* x     = (const float*)d_in[0];
  const float* w_qkv = (const float*)d_in[1];
  const float* b_qkv = (const float*)d_in[2];
  const float* w_o   = (const float*)d_in[3];
  const float* b_o   = (const float*)d_in[4];

  const size_t NOUT = (size_t)BATCH * T_SEQ * D_MODEL;  // 4,194,304
  float* out  = (float*)d_out;
  float* kout = out + NOUT;      // [B, H, T, dk]
  float* vout = kout + NOUT;     // [B, H, T, dk]

  float* qws = (float*)d_ws;     // [B, H, T, dk] fp32, 16 MB
  float* ctx = qws + NOUT;       // [B, T, D]     fp32, 16 MB

  qkv_proj_kernel<<<dim3(3 * D_MODEL / 512, BATCH * T_SEQ / 16), 256, 0, stream>>>(
      x, w_qkv, b_qkv, qws, kout, vout);
  attn_kernel<<<dim3(T_SEQ / 64, BATCH * HEADS), 128, 0, stream>>>(
      qws, kout, vout, ctx);
  out_proj_kernel<<<dim3(D_MODEL / 512, BATCH * T_SEQ / 16), 256, 0, stream>>>(
      ctx, w_o, b_o, out);
}